// MessagePassingLayer_48335561949417
// MI455X (gfx1250) — compile-verified
//
#include <hip/hip_runtime.h>
#include <math.h>

typedef __attribute__((ext_vector_type(2))) float v2f;
typedef __attribute__((ext_vector_type(8))) float v8f;

constexpr int   Bc = 16, Nc = 512, Hc = 160, BINSc = 20, AHc = 40;
constexpr float CUTOFFc = 10.0f, GAMMAc = 2.0f, LN_EPSc = 1e-5f;
constexpr float RBF_STEP = CUTOFFc / (BINSc - 1);   // linspace(0,10,20) spacing
constexpr float L2E = 1.4426950408889634f;          // log2(e)

// Branch-free exp for non-positive-ish args (underflow-to-0 is correct here).
#if __has_builtin(__builtin_amdgcn_exp2f)
__device__ __forceinline__ float exp2_raw(float x) { return __builtin_amdgcn_exp2f(x); }
#else
__device__ __forceinline__ float exp2_raw(float x) { return exp2f(x); }
#endif
__device__ __forceinline__ float exp_fast(float x) { return exp2_raw(x * L2E); }

// Branch-free tanh: native v_tanh_f32 on gfx1250 if available.
__device__ __forceinline__ float tanh_fast(float x) {
#if __has_builtin(__builtin_amdgcn_tanhf)
    return __builtin_amdgcn_tanhf(x);
#else
    float e = exp2_raw(-2.0f * L2E * fabsf(x));        // e^(-2|x|)
#if __has_builtin(__builtin_amdgcn_rcpf)
    float t = (1.0f - e) * __builtin_amdgcn_rcpf(1.0f + e);
#else
    float t = (1.0f - e) / (1.0f + e);
#endif
    return copysignf(t, x);
#endif
}

__device__ __forceinline__ float gelu_t(float x) {
    const float k = 0.7978845608028654f;               // sqrt(2/pi)
    float x3 = x * x * x;
    return 0.5f * x * (1.0f + tanh_fast(k * (x + 0.044715f * x3)));
}

__device__ __forceinline__ v8f wmma4(v2f a, v2f b, v8f c) {
    // D = A(16x4,f32) * B(4x16,f32) + C(16x16,f32)
    return __builtin_amdgcn_wmma_f32_16x16x4_f32(false, a, false, b, (short)0, c, false, false);
}

// ---------------- Kernel 1: transformed = gelu(h @ wm + bm) ----------------
// GEMM M=B*N=8192, K=160, N=160. 512*10 = 5120 wave tiles, 8 waves/block.
__global__ __launch_bounds__(256) void k_transform(const float* __restrict__ h,
                                                   const float* __restrict__ wm,
                                                   const float* __restrict__ bm,
                                                   float* __restrict__ T) {
    const int wave = threadIdx.x >> 5, lane = threadIdx.x & 31;
    const int col = lane & 15, koff = (lane < 16) ? 0 : 2;
    const int id = blockIdx.x * 8 + wave;
    const int ntile = id % 10, mtile = id / 10;
    const int n0 = ntile * 16, m0 = mtile * 16;

    v8f c;
    const float bias = bm[n0 + col];
#pragma unroll
    for (int r = 0; r < 8; ++r) c[r] = bias;

    const float* arow = h + (size_t)(m0 + col) * Hc;
    for (int k0 = 0; k0 < Hc; k0 += 4) {
        const int kb = k0 + koff;
        float2 av = *(const float2*)(arow + kb);
        v2f a; a.x = av.x; a.y = av.y;
        v2f b; b.x = wm[(size_t)kb * Hc + n0 + col];
               b.y = wm[(size_t)(kb + 1) * Hc + n0 + col];
        c = wmma4(a, b, c);
    }
    const int rb = m0 + ((lane < 16) ? 0 : 8);
#pragma unroll
    for (int r = 0; r < 8; ++r)
        T[(size_t)(rb + r) * Hc + n0 + col] = gelu_t(c[r]);
}

// ---------------- Kernel 2: attention scores (WMMA pair-MLP) + softmax -----
// One block (256 threads, 8 waves) per (b, i). Each wave handles 4 groups of
// 16 j's: rbf built in registers as the A matrix, w1a as B from LDS.
__global__ __launch_bounds__(256) void k_attn(const float* __restrict__ pos,
                                              const unsigned char* __restrict__ mask,
                                              const float* __restrict__ w1a,
                                              const float* __restrict__ b1a,
                                              const float* __restrict__ w2a,
                                              const float* __restrict__ b2a,
                                              float* __restrict__ attn) {
    __shared__ float s[Nc];
    __shared__ float red[256];
    __shared__ float w1s[BINSc * 48];   // zero-padded 20 x 48
    __shared__ float b1s[48];
    __shared__ float w2s[48];

    const int b = blockIdx.x / Nc, i = blockIdx.x % Nc;
    const int t = threadIdx.x;

    for (int idx = t; idx < BINSc * 48; idx += 256) {
        int k = idx / 48, n = idx % 48;
        w1s[idx] = (n < AHc) ? w1a[k * AHc + n] : 0.0f;
    }
    if (t < 48) {
        b1s[t] = (t < AHc) ? b1a[t] : 0.0f;
        w2s[t] = (t < AHc) ? w2a[t] : 0.0f;
    }
    __syncthreads();

    const int wave = t >> 5, lane = t & 31;
    const int col = lane & 15, koff = (lane < 16) ? 0 : 2;
    const float pix = pos[(size_t)(b * Nc + i) * 3 + 0];
    const float piy = pos[(size_t)(b * Nc + i) * 3 + 1];
    const float piz = pos[(size_t)(b * Nc + i) * 3 + 2];
    const float b2v = b2a[0];
    const float negGL2E = -GAMMAc * L2E;   // rbf = exp2(negGL2E * t^2)

    for (int q = 0; q < 4; ++q) {
        const int jg = wave * 4 + q;
        const int jj = jg * 16 + col;
        float dx = pix - pos[(size_t)(b * Nc + jj) * 3 + 0];
        float dy = piy - pos[(size_t)(b * Nc + jj) * 3 + 1];
        float dz = piz - pos[(size_t)(b * Nc + jj) * 3 + 2];
        float d = sqrtf(fmaxf(dx * dx + dy * dy + dz * dz, 1e-12f));

        v8f acc;
#pragma unroll
        for (int r = 0; r < 8; ++r) acc[r] = 0.0f;

        for (int tl = 0; tl < 3; ++tl) {           // hidden tiles (40 -> 3x16)
            v8f c;
            const float bias = b1s[tl * 16 + col];
#pragma unroll
            for (int r = 0; r < 8; ++r) c[r] = bias;
#pragma unroll
            for (int s5 = 0; s5 < 5; ++s5) {       // K = 20 -> 5 steps of 4
                const int kb = s5 * 4 + koff;
                float t0 = d - (float)kb * RBF_STEP;
                float t1 = d - (float)(kb + 1) * RBF_STEP;
                v2f a; a.x = exp2_raw(negGL2E * t0 * t0);
                       a.y = exp2_raw(negGL2E * t1 * t1);
                v2f bb; bb.x = w1s[kb * 48 + tl * 16 + col];
                        bb.y = w1s[(kb + 1) * 48 + tl * 16 + col];
                c = wmma4(a, bb, c);
            }
            const float w2v = w2s[tl * 16 + col];
#pragma unroll
            for (int r = 0; r < 8; ++r) acc[r] += gelu_t(c[r]) * w2v;
        }
        // reduce over the 16-lane half (rows r in lanes 0-15, r+8 in 16-31)
#pragma unroll
        for (int r = 0; r < 8; ++r) {
            float v = acc[r];
            v += __shfl_xor(v, 1); v += __shfl_xor(v, 2);
            v += __shfl_xor(v, 4); v += __shfl_xor(v, 8);
            acc[r] = v;
        }
        if (lane == 0) {
#pragma unroll
            for (int r = 0; r < 8; ++r) s[jg * 16 + r] = acc[r] + b2v;
        } else if (lane == 16) {
#pragma unroll
            for (int r = 0; r < 8; ++r) s[jg * 16 + 8 + r] = acc[r] + b2v;
        }
    }
    __syncthreads();

    // neighbor mask + softmax over j
    const unsigned char mi = mask[(size_t)b * Nc + i];
    float lmax = -3.4e38f;
    for (int j = t; j < Nc; j += 256) {
        float dx = pix - pos[(size_t)(b * Nc + j) * 3 + 0];
        float dy = piy - pos[(size_t)(b * Nc + j) * 3 + 1];
        float dz = piz - pos[(size_t)(b * Nc + j) * 3 + 2];
        float d = sqrtf(fmaxf(dx * dx + dy * dy + dz * dz, 1e-12f));
        bool nbr = (d < CUTOFFc) && (d > 0.01f) && mi && mask[(size_t)b * Nc + j];
        float v = nbr ? s[j] : -60000.0f;
        s[j] = v;
        lmax = fmaxf(lmax, v);
    }
    red[t] = lmax; __syncthreads();
    for (int off = 128; off > 0; off >>= 1) {
        if (t < off) red[t] = fmaxf(red[t], red[t + off]);
        __syncthreads();
    }
    const float gmax = red[0]; __syncthreads();

    float lsum = 0.0f;
    for (int j = t; j < Nc; j += 256) {
        float e = exp_fast(s[j] - gmax);   // arg <= 0: underflow-to-0 correct
        s[j] = e; lsum += e;
    }
    red[t] = lsum; __syncthreads();
    for (int off = 128; off > 0; off >>= 1) {
        if (t < off) red[t] += red[t + off];
        __syncthreads();
    }
    const float inv = 1.0f / red[0];
    for (int j = t; j < Nc; j += 256)
        attn[((size_t)b * Nc + i) * Nc + j] = s[j] * inv;
}

// ---------------- Kernel 3: msg = attn @ transformed (batched GEMM) --------
// One block per (b, i-tile): 320 threads = 10 waves = 10 h-tiles (H=160).
// The shared 16x512 attn A-tile is staged into LDS once via the gfx1250
// async global->LDS path, then each wave reads it conflict-free (row pad 4).
__global__ __launch_bounds__(320) void k_aggregate(const float* __restrict__ attn,
                                                   const float* __restrict__ T,
                                                   float* __restrict__ msg) {
    __shared__ float As[16 * 516];       // 512 + 4 pad per row, 33 KB
    const int b = blockIdx.x / 32;
    const int i0 = (blockIdx.x % 32) * 16;
    const int t = threadIdx.x;

    // Async-stage A tile: 16 rows x 512 floats = 2048 x 16B chunks.
    const float* abase = attn + ((size_t)b * Nc + i0) * Nc;
    for (int cidx = t; cidx < 2048; cidx += 320) {
        const int row = cidx >> 7;              // 128 chunks per row
        const int cc = cidx & 127;
        unsigned lds_off = (unsigned)((row * 516 + cc * 4) * 4);
        const float* g = abase + (size_t)row * Nc + cc * 4;
        asm volatile("global_load_async_to_lds_b128 %0, %1, off"
                     :: "v"(lds_off), "v"(g) : "memory");
    }
    asm volatile("s_wait_asynccnt 0x0" ::: "memory");
    __syncthreads();

    const int wave = t >> 5, lane = t & 31;
    const int col = lane & 15, koff = (lane < 16) ? 0 : 2;
    const int h0 = wave * 16;                   // 10 waves -> h-tiles 0..9

    const float* arowS = &As[(lane & 15) * 516];
    const float* Tb = T + (size_t)b * Nc * Hc + h0 + col;

    v8f c;
#pragma unroll
    for (int r = 0; r < 8; ++r) c[r] = 0.0f;

    for (int k0 = 0; k0 < Nc; k0 += 4) {
        const int kb = k0 + koff;
        float2 av = *(const float2*)(arowS + kb);    // ds_load_b64, no conflicts
        v2f a; a.x = av.x; a.y = av.y;
        v2f bb; bb.x = Tb[(size_t)kb * Hc];
                bb.y = Tb[(size_t)(kb + 1) * Hc];
        c = wmma4(a, bb, c);
    }
    const int rb = i0 + ((lane < 16) ? 0 : 8);
#pragma unroll
    for (int r = 0; r < 8; ++r)
        msg[((size_t)b * Nc + rb + r) * Hc + h0 + col] = c[r];
}

// ---------------- Kernel 4: out = LayerNorm(h + msg) -----------------------
__global__ __launch_bounds__(256) void k_layernorm(const float* __restrict__ h,
                                                   const float* __restrict__ msg,
                                                   const float* __restrict__ gamma,
                                                   const float* __restrict__ beta,
                                                   float* __restrict__ out) {
    const int wave = threadIdx.x >> 5, lane = threadIdx.x & 31;
    const int row = blockIdx.x * 8 + wave;
    const float* hr = h + (size_t)row * Hc;
    const float* mr = msg + (size_t)row * Hc;

    float x[5], sum = 0.0f, sq = 0.0f;
#pragma unroll
    for (int e = 0; e < 5; ++e) {
        int idx = lane + 32 * e;
        float v = hr[idx] + mr[idx];
        x[e] = v; sum += v; sq += v * v;
    }
#pragma unroll
    for (int m = 1; m < 32; m <<= 1) {
        sum += __shfl_xor(sum, m);
        sq  += __shfl_xor(sq, m);
    }
    const float mu = sum * (1.0f / Hc);
    const float var = sq * (1.0f / Hc) - mu * mu;
    const float rs = rsqrtf(var + LN_EPSc);
#pragma unroll
    for (int e = 0; e < 5; ++e) {
        int idx = lane + 32 * e;
        out[(size_t)row * Hc + idx] = (x[e] - mu) * rs * gamma[idx] + beta[idx];
    }
}

extern "C" void kernel_launch(void* const* d_in, const int* in_sizes, int n_in,
                              void* d_out, int out_size, void* d_ws, size_t ws_size,
                              hipStream_t stream) {
    const float*         h     = (const float*)d_in[0];
    const float*         pos   = (const float*)d_in[1];
    const unsigned char* mask  = (const unsigned char*)d_in[2];
    const float*         w1a   = (const float*)d_in[3];
    const float*         b1a   = (const float*)d_in[4];
    const float*         w2a   = (const float*)d_in[5];
    const float*         b2a   = (const float*)d_in[6];
    const float*         wm    = (const float*)d_in[7];
    const float*         bm    = (const float*)d_in[8];
    const float*         gamma = (const float*)d_in[9];
    const float*         beta  = (const float*)d_in[10];

    float* ws   = (float*)d_ws;
    float* T    = ws;                                  // [16,512,160] = 5.2 MB
    float* attn = T + (size_t)Bc * Nc * Hc;            // [16,512,512] = 16.8 MB
    float* msg  = attn + (size_t)Bc * Nc * Nc;         // [16,512,160] = 5.2 MB

    k_transform<<<640, 256, 0, stream>>>(h, wm, bm, T);
    k_attn<<<Bc * Nc, 256, 0, stream>>>(pos, mask, w1a, b1a, w2a, b2a, attn);
    k_aggregate<<<512, 320, 0, stream>>>(attn, T, msg);
    k_layernorm<<<1024, 256, 0, stream>>>(h, msg, gamma, beta, (float*)d_out);
}